// S4Layer_10479720202533
// MI455X (gfx1250) — compile-verified
//
#include <hip/hip_runtime.h>
#include <hip/hip_bf16.h>

typedef __attribute__((ext_vector_type(2))) float v2f;
typedef __attribute__((ext_vector_type(8))) float v8f;

#define NST 64    // state dim
#define TCH 64    // chunk length
#define LSEQ 4096
#define DMOD 512
#define NSEG 16   // segments along time
#define SCH 4     // chunks per segment (NSEG*SCH*TCH == LSEQ)

// ws layout (floats):
//   0     TKt[q][p] = K[p-q] (p>=q)
//   4096  Mt [q][n] = (A^{63-q} b)[n]
//   8192  Et [n][p] = (c A^{p+1})[n]
//   12288 Pt [m][n] = (A^64)[n][m]
//   16384 PSt[m][n] = (A^256)[n][m]          (= Pt^4, segment transition)
//   20480 ST[tile(128)][seg(16)][16][64]     (segment states, 8 MB)
#define WS_ST 20480

// ---------------------------------------------------------------------------
// Kernel 1: one 64-thread block builds the 5 chunk/segment matrices.
// ---------------------------------------------------------------------------
__global__ __launch_bounds__(64) void s4_precompute(const float* __restrict__ A_,
                                                    const float* __restrict__ B_,
                                                    const float* __restrict__ C_,
                                                    float* __restrict__ ws) {
  __shared__ float As[NST * NST];
  __shared__ float Sq[NST * NST];
  __shared__ float Sq2[NST * NST];
  __shared__ float vbuf[NST][NST];      // vbuf[t][i] = (A^t b)[i]
  __shared__ float wbuf[NST + 1][NST];  // wbuf[p][j] = (c A^p)[j], p = 1..64
  __shared__ float cb[NST];
  __shared__ float Kk[NST];

  const int tid = threadIdx.x;  // 0..63
#pragma clang loop unroll(disable)
  for (int j = 0; j < NST; ++j) As[tid * NST + j] = A_[tid * NST + j];
  cb[tid] = C_[tid];
  vbuf[0][tid] = B_[tid];
  __syncthreads();

  // v_t = A^t b
#pragma clang loop unroll(disable)
  for (int t = 1; t < NST; ++t) {
    float s = 0.f;
#pragma clang loop unroll(disable)
    for (int j = 0; j < NST; ++j) s += As[tid * NST + j] * vbuf[t - 1][j];
    vbuf[t][tid] = s;
    __syncthreads();
  }

  // w_p = c A^p, p = 1..64
#pragma clang loop unroll(disable)
  for (int p = 1; p <= NST; ++p) {
    const float* prev = (p == 1) ? cb : wbuf[p - 1];
    float s = 0.f;
#pragma clang loop unroll(disable)
    for (int i = 0; i < NST; ++i) s += prev[i] * As[i * NST + tid];
    wbuf[p][tid] = s;
    __syncthreads();
  }

  // K[t] = c . v_t
  {
    float s = 0.f;
#pragma clang loop unroll(disable)
    for (int i = 0; i < NST; ++i) s += cb[i] * vbuf[tid][i];
    Kk[tid] = s;
  }
  __syncthreads();

  // A^64 by 6 LDS squarings (ping-pong)
  const float* src = As;
  float* dst = Sq2;
#pragma clang loop unroll(disable)
  for (int it = 0; it < 6; ++it) {
#pragma clang loop unroll(disable)
    for (int j = 0; j < NST; ++j) {
      float s = 0.f;
#pragma clang loop unroll(disable)
      for (int k = 0; k < NST; ++k) s += src[tid * NST + k] * src[k * NST + j];
      dst[tid * NST + j] = s;
    }
    __syncthreads();
    src = dst;
    dst = (src == Sq2) ? Sq : Sq2;
  }
  const float* A64 = src;

  float* TKt = ws;
  float* Mt  = ws + 4096;
  float* Et  = ws + 8192;
  float* Pt  = ws + 12288;
#pragma clang loop unroll(disable)
  for (int j = 0; j < NST; ++j) {
    TKt[tid * NST + j] = (j >= tid) ? Kk[j - tid] : 0.f;  // q=tid, p=j
    Mt [tid * NST + j] = vbuf[NST - 1 - tid][j];          // q=tid, n=j
    Et [tid * NST + j] = wbuf[j + 1][tid];                // n=tid, p=j
    Pt [tid * NST + j] = A64[j * NST + tid];              // m=tid, n=j
  }
  __syncthreads();  // all A64 reads done before it gets overwritten

  // 2 more squarings: A^128, A^256
#pragma clang loop unroll(disable)
  for (int it = 0; it < 2; ++it) {
#pragma clang loop unroll(disable)
    for (int j = 0; j < NST; ++j) {
      float s = 0.f;
#pragma clang loop unroll(disable)
      for (int k = 0; k < NST; ++k) s += src[tid * NST + k] * src[k * NST + j];
      dst[tid * NST + j] = s;
    }
    __syncthreads();
    src = dst;
    dst = (src == Sq2) ? Sq : Sq2;
  }
  const float* A256 = src;
  float* PSt = ws + 16384;
#pragma clang loop unroll(disable)
  for (int j = 0; j < NST; ++j) {
    PSt[tid * NST + j] = A256[j * NST + tid];  // m=tid, n=j
  }
}

// ---------------------------------------------------------------------------
// Phase A: 2048 waves (128 seq-tiles x 16 segments). Each wave runs its
// 4-chunk segment from zero state, only the state recurrence
//   X <- U @ Mt + X @ Pt
// and writes the segment-generated state G_seg to ST[tile][seg].
// ---------------------------------------------------------------------------
__global__ __launch_bounds__(32)
__attribute__((amdgpu_num_vgpr(256)))
void s4_segstate(const float* __restrict__ u, float* __restrict__ ws) {
  __shared__ float sM[2 * 4096];  // Mt | Pt
  __shared__ float sX[2][1024];   // ping-pong state [seq][state]

  const int tid = threadIdx.x;
  {
    const float4* m4 = (const float4*)(ws + 4096);   // Mt
    const float4* p4 = (const float4*)(ws + 12288);  // Pt
    float4* d0 = (float4*)sM;
    float4* d1 = (float4*)(sM + 4096);
    for (int i = tid; i < 1024; i += 32) d0[i] = m4[i];
    for (int i = tid; i < 1024; i += 32) d1[i] = p4[i];
    for (int i = tid; i < 1024; i += 32) sX[0][i] = 0.f;
  }
  __syncthreads();

  const int tile = blockIdx.x >> 4;
  const int seg  = blockIdx.x & 15;
  const int s0 = tile * 16;
  const int bb = s0 >> 9, dm0 = s0 & 511;
  const int row = tid & 15, hi = tid >> 4;
  const float* ub0 = u + (long)bb * LSEQ * DMOD + dm0 + row + (long)(2 * hi) * DMOD;

  for (int i = 0; i < SCH; ++i) {
    const int c = seg * SCH + i;
    const float* Xr = sX[i & 1];
    float* Xw = (float*)sX[(i & 1) ^ 1];
    const float* up = ub0 + (long)c * TCH * DMOD;
    v2f uf[16];
#pragma unroll
    for (int kk = 0; kk < 16; ++kk) {
      uf[kk].x = up[kk * 4 * DMOD];
      uf[kk].y = up[kk * 4 * DMOD + DMOD];
    }
    const v2f* xr = (const v2f*)(Xr + row * 64 + 2 * hi);
    v2f xin[16];
#pragma unroll
    for (int kk = 0; kk < 16; ++kk) xin[kk] = xr[kk * 2];

#pragma unroll
    for (int nt = 0; nt < 4; ++nt) {
      v8f ax = {0.f, 0.f, 0.f, 0.f, 0.f, 0.f, 0.f, 0.f};
      const int col = nt * 16 + row;
      const float* pM = sM + col;
      const float* pP = sM + 4096 + col;
#pragma unroll
      for (int kk = 0; kk < 16; ++kk) {
        const int k0 = 4 * kk + 2 * hi;
        v2f bm; bm.x = pM[k0 * 64]; bm.y = pM[(k0 + 1) * 64];
        ax = __builtin_amdgcn_wmma_f32_16x16x4_f32(false, uf[kk], false, bm,
                                                   (short)0, ax, false, false);
        v2f bp; bp.x = pP[k0 * 64]; bp.y = pP[(k0 + 1) * 64];
        ax = __builtin_amdgcn_wmma_f32_16x16x4_f32(false, xin[kk], false, bp,
                                                   (short)0, ax, false, false);
      }
      float* xw = Xw + (8 * hi) * 64 + col;
#pragma unroll
      for (int r = 0; r < 8; ++r) xw[r * 64] = ax[r];
    }
    __syncthreads();
  }
  // Final state is back in sX[0] after 4 ping-pongs; write G_seg.
  float* st = ws + WS_ST + (long)tile * (NSEG * 1024) + seg * 1024;
  const float4* xp = (const float4*)sX[0];
  float4* gp = (float4*)st;
  for (int i = tid; i < 256; i += 32) gp[i] = xp[i];
}

// ---------------------------------------------------------------------------
// Phase B: 128 waves (one per seq-tile). Segment-level scan
//   E_{g+1} = E_g @ PSt + G_g,  E_0 = 0,
// storing E_{g+1} into slot g (safe: store data depends on the loaded G_g
// through the WMMA chain, so loads complete first). Phase C reads the
// entering state of segment s from slot s-1 (s=0 starts from zero).
// ---------------------------------------------------------------------------
__global__ __launch_bounds__(32) void s4_combine(float* __restrict__ ws) {
  __shared__ float sP[4096];     // PSt
  __shared__ float sE[2][1024];  // ping-pong E

  const int tid = threadIdx.x;
  {
    const float4* p4 = (const float4*)(ws + 16384);
    float4* d4 = (float4*)sP;
    for (int i = tid; i < 1024; i += 32) d4[i] = p4[i];
    for (int i = tid; i < 1024; i += 32) sE[0][i] = 0.f;
  }
  __syncthreads();

  const int row = tid & 15, hi = tid >> 4;
  float* stt = ws + WS_ST + (long)blockIdx.x * (NSEG * 1024);

#pragma clang loop unroll(disable)
  for (int g = 0; g < NSEG - 1; ++g) {
    float* slot = stt + g * 1024;
    const float* Er = sE[g & 1];
    float* Ew = (float*)sE[(g & 1) ^ 1];
    const v2f* er = (const v2f*)(Er + row * 64 + 2 * hi);
    v2f ef[16];
#pragma unroll
    for (int kk = 0; kk < 16; ++kk) ef[kk] = er[kk * 2];

#pragma unroll
    for (int nt = 0; nt < 4; ++nt) {
      const int col = nt * 16 + row;
      v8f ax;  // accumulator initialized from G_g (D layout)
#pragma unroll
      for (int r = 0; r < 8; ++r) ax[r] = slot[(r + 8 * hi) * 64 + col];
      const float* pP = sP + col;
#pragma unroll
      for (int kk = 0; kk < 16; ++kk) {
        const int k0 = 4 * kk + 2 * hi;
        v2f bp; bp.x = pP[k0 * 64]; bp.y = pP[(k0 + 1) * 64];
        ax = __builtin_amdgcn_wmma_f32_16x16x4_f32(false, ef[kk], false, bp,
                                                   (short)0, ax, false, false);
      }
#pragma unroll
      for (int r = 0; r < 8; ++r) slot[(r + 8 * hi) * 64 + col] = ax[r];
      float* ew = Ew + (8 * hi) * 64 + col;
#pragma unroll
      for (int r = 0; r < 8; ++r) ew[r * 64] = ax[r];
    }
    __syncthreads();
  }
}

// ---------------------------------------------------------------------------
// Phase C: 2048 waves. Each wave loads its entering state E_seg and replays
// its 4 chunks with the full output GEMMs:
//   Y     = U @ TKt + Xin @ Et
//   Xnext = U @ Mt  + Xin @ Pt
// ---------------------------------------------------------------------------
__global__ __launch_bounds__(32)
__attribute__((amdgpu_num_vgpr(256)))
void s4_outputs(const float* __restrict__ u, const float* __restrict__ ws,
                float* __restrict__ y) {
  __shared__ float sMat[4 * 4096];  // TKt | Mt | Et | Pt
  __shared__ float sX[2][1024];

  const int tid = threadIdx.x;
  {
    const float4* s4p = (const float4*)ws;
    float4* d4 = (float4*)sMat;
    for (int i = tid; i < 4096; i += 32) d4[i] = s4p[i];
  }
  const int tile = blockIdx.x >> 4;
  const int seg  = blockIdx.x & 15;
  if (seg == 0) {
    for (int i = tid; i < 1024; i += 32) sX[0][i] = 0.f;
  } else {
    const float4* ep =
        (const float4*)(ws + WS_ST + (long)tile * (NSEG * 1024) + (seg - 1) * 1024);
    float4* xp = (float4*)sX[0];
    for (int i = tid; i < 256; i += 32) xp[i] = ep[i];
  }
  __syncthreads();

  const int s0 = tile * 16;
  const int bb = s0 >> 9, dm0 = s0 & 511;
  const int row = tid & 15, hi = tid >> 4;
  const float* ub0 = u + (long)bb * LSEQ * DMOD + dm0 + row + (long)(2 * hi) * DMOD;
  float* yb0 = y + (long)bb * LSEQ * DMOD + dm0 + 8 * hi;

  for (int i = 0; i < SCH; ++i) {
    const int c = seg * SCH + i;
    const float* Xr = sX[i & 1];
    float* Xw = (float*)sX[(i & 1) ^ 1];
    const float* up = ub0 + (long)c * TCH * DMOD;
    v2f uf[16];
#pragma unroll
    for (int kk = 0; kk < 16; ++kk) {
      uf[kk].x = up[kk * 4 * DMOD];
      uf[kk].y = up[kk * 4 * DMOD + DMOD];
    }
    const v2f* xr = (const v2f*)(Xr + row * 64 + 2 * hi);
    v2f xin[16];
#pragma unroll
    for (int kk = 0; kk < 16; ++kk) xin[kk] = xr[kk * 2];

#pragma unroll
    for (int nt = 0; nt < 4; ++nt) {
      v8f ay = {0.f, 0.f, 0.f, 0.f, 0.f, 0.f, 0.f, 0.f};
      v8f ax = {0.f, 0.f, 0.f, 0.f, 0.f, 0.f, 0.f, 0.f};
      const int col = nt * 16 + row;
      const float* pTK = sMat + col;
      const float* pMt = sMat + 4096 + col;
      const float* pEt = sMat + 8192 + col;
      const float* pPt = sMat + 12288 + col;
#pragma unroll
      for (int kk = 0; kk < 16; ++kk) {
        const int k0 = 4 * kk + 2 * hi;
        v2f bt; bt.x = pTK[k0 * 64]; bt.y = pTK[(k0 + 1) * 64];
        ay = __builtin_amdgcn_wmma_f32_16x16x4_f32(false, uf[kk], false, bt,
                                                   (short)0, ay, false, false);
        v2f bm; bm.x = pMt[k0 * 64]; bm.y = pMt[(k0 + 1) * 64];
        ax = __builtin_amdgcn_wmma_f32_16x16x4_f32(false, uf[kk], false, bm,
                                                   (short)0, ax, false, false);
        v2f be; be.x = pEt[k0 * 64]; be.y = pEt[(k0 + 1) * 64];
        ay = __builtin_amdgcn_wmma_f32_16x16x4_f32(false, xin[kk], false, be,
                                                   (short)0, ay, false, false);
        v2f bp; bp.x = pPt[k0 * 64]; bp.y = pPt[(k0 + 1) * 64];
        ax = __builtin_amdgcn_wmma_f32_16x16x4_f32(false, xin[kk], false, bp,
                                                   (short)0, ax, false, false);
      }
      // Y tile: two aligned b128 stores.
      float4* yp = (float4*)(yb0 + (long)(c * TCH + col) * DMOD);
      float4 lo, hh;
      lo.x = ay[0]; lo.y = ay[1]; lo.z = ay[2]; lo.w = ay[3];
      hh.x = ay[4]; hh.y = ay[5]; hh.z = ay[6]; hh.w = ay[7];
      yp[0] = lo;
      yp[1] = hh;
      float* xw = Xw + (8 * hi) * 64 + col;
#pragma unroll
      for (int r = 0; r < 8; ++r) xw[r * 64] = ax[r];
    }
    __syncthreads();
  }
}

// ---------------------------------------------------------------------------
extern "C" void kernel_launch(void* const* d_in, const int* in_sizes, int n_in,
                              void* d_out, int out_size, void* d_ws, size_t ws_size,
                              hipStream_t stream) {
  const float* u  = (const float*)d_in[0];
  const float* A_ = (const float*)d_in[1];
  const float* B_ = (const float*)d_in[2];
  const float* C_ = (const float*)d_in[3];
  float* y  = (float*)d_out;
  float* ws = (float*)d_ws;  // needs (20480 + 128*16*1024) floats ~ 8.5 MB

  s4_precompute<<<1, 64, 0, stream>>>(A_, B_, C_, ws);
  s4_segstate<<<128 * NSEG, 32, 0, stream>>>(u, ws);
  s4_combine<<<128, 32, 0, stream>>>(ws);
  s4_outputs<<<128 * NSEG, 32, 0, stream>>>(u, ws, y);
}